// FeedForwardQuantum_65481071401855
// MI455X (gfx1250) — compile-verified
//
#include <hip/hip_runtime.h>
#include <math.h>

typedef float v2f __attribute__((ext_vector_type(2)));
typedef float v4f __attribute__((ext_vector_type(4)));
typedef float v8f __attribute__((ext_vector_type(8)));

#if defined(__HIP_DEVICE_COMPILE__)
#if !__has_builtin(__builtin_amdgcn_wmma_f32_16x16x4_f32)
#error "device: missing __builtin_amdgcn_wmma_f32_16x16x4_f32"
#endif
#endif

#define EMBED 1024
#define NW 8
#define WAVES_PER_BLOCK 8
#define QSTRIDE 16  // floats; 64B rows -> b128-aligned A-fragment loads

__global__ __launch_bounds__(256) void ffq_wmma_kernel(
    const float* __restrict__ x,
    const float* __restrict__ W1,
    const float* __restrict__ b1,
    const float* __restrict__ phi,
    const float* __restrict__ W2,
    const float* __restrict__ b2,
    float* __restrict__ out)
{
#if defined(__HIP_DEVICE_COMPILE__)
    __shared__ __align__(16) float qsh[WAVES_PER_BLOCK][16][QSTRIDE];

    const int tid  = threadIdx.x;
    const int wave = tid >> 5;
    const int lane = tid & 31;
    const int hl   = lane >> 4;   // wave half: selects K'{0,1} vs K'{2,3} (A/B) or M 0-7 vs 8-15 (C/D)
    const int lm   = lane & 15;   // M index (A) or N index (B/C/D)

    const long subtile = (long)blockIdx.x * WAVES_PER_BLOCK + wave; // 16 tokens each
    const long tok0    = subtile * 16;

    // ---------------- GEMM1: inter[16x16] = x_tile[16x1024] * W1^T ---------------------
    // K is consumed in the permuted order {8i,8i+1,8i+4,8i+5} / {8i+2,8i+3,8i+6,8i+7} so
    // each lane's single b128 load feeds two WMMAs (A and B use the same permutation, so
    // the dot product is unchanged). Lanes n>=8 duplicate W1 row n&7: result columns
    // 8..15 are finite garbage that is never consumed (GEMM2 reads only q cols 0..7).
    const float* xp = x  + (tok0 + lm) * EMBED + 4 * hl;
    const float* wp = W1 + (long)(lm & 7) * EMBED + 4 * hl;

    v8f c1 = {0.f, 0.f, 0.f, 0.f, 0.f, 0.f, 0.f, 0.f};
    #pragma unroll 8
    for (int k = 0; k < EMBED; k += 8) {
        v4f xa = *(const v4f*)(xp + k);
        v4f wb = *(const v4f*)(wp + k);
        v2f a0 = {xa.x, xa.y}, b0 = {wb.x, wb.y};
        v2f a1 = {xa.z, xa.w}, b1f = {wb.z, wb.w};
        c1 = __builtin_amdgcn_wmma_f32_16x16x4_f32(
                 false, a0, false, b0, (short)0, c1, false, false);
        c1 = __builtin_amdgcn_wmma_f32_16x16x4_f32(
                 false, a1, false, b1f, (short)0, c1, false, false);
    }

    // ---------------- epilogue: q = cos(relu(inter + b1) + phi) -------------------------
    const float b1v = b1[lm & 7];
    const float phv = phi[lm & 7];

    // D layout: lane, vgpr r -> D[M = r + 8*hl][N = lm]
    // Arguments are O(1..30) rad, far inside the v_cos_f32 fast range -> __cosf is exact
    // enough and avoids the libm Payne-Hanek slow path (big VALU blob + VGPR pressure).
    #pragma unroll
    for (int r = 0; r < 8; ++r) {
        float v = c1[r] + b1v;
        v = fmaxf(v, 0.f) + phv;
        qsh[wave][r + 8 * hl][lm] = __cosf(v);
    }
    // wave-private LDS region; compiler inserts the s_wait_dscnt for the RAW below.

    // One b128 per lane provides both A-fragments of q[16x8] (permuted K {0,1,4,5}/{2,3,6,7})
    v4f q4 = *(const v4f*)&qsh[wave][lm][4 * hl];
    v2f a_lo = {q4.x, q4.y};   // logical K 0,1 (h=0) / 4,5 (h=1)
    v2f a_hi = {q4.z, q4.w};   // logical K 2,3 (h=0) / 6,7 (h=1)

    // ---------------- GEMM2: out_tile[16x1024] = q[16x8] * W2^T + b2 --------------------
    const long obase = tok0 * EMBED;
    #pragma unroll 2
    for (int e0 = 0; e0 < EMBED; e0 += 16) {
        // One b128 of the W2 row per lane feeds both WMMAs (same K permutation as A).
        v4f w24 = *(const v4f*)(W2 + (long)(e0 + lm) * NW + 4 * hl);
        v2f bw1 = {w24.x, w24.y};
        v2f bw2 = {w24.z, w24.w};

        v8f c2 = {0.f, 0.f, 0.f, 0.f, 0.f, 0.f, 0.f, 0.f};
        c2 = __builtin_amdgcn_wmma_f32_16x16x4_f32(
                 false, a_lo, false, bw1, (short)0, c2, false, false);
        c2 = __builtin_amdgcn_wmma_f32_16x16x4_f32(
                 false, a_hi, false, bw2, (short)0, c2, false, false);

        float b2v = b2[e0 + lm];
        #pragma unroll
        for (int r = 0; r < 8; ++r) {
            out[obase + (long)(r + 8 * hl) * EMBED + e0 + lm] = c2[r] + b2v;
        }
    }
#endif  // __HIP_DEVICE_COMPILE__
}

extern "C" void kernel_launch(void* const* d_in, const int* in_sizes, int n_in,
                              void* d_out, int out_size, void* d_ws, size_t ws_size,
                              hipStream_t stream) {
    const float* x   = (const float*)d_in[0];
    const float* W1  = (const float*)d_in[1];
    const float* b1  = (const float*)d_in[2];
    const float* phi = (const float*)d_in[3];
    const float* W2  = (const float*)d_in[4];
    const float* b2  = (const float*)d_in[5];
    float* out = (float*)d_out;

    const int tokens   = in_sizes[0] / EMBED;        // 32768
    const int subtiles = tokens / 16;                // 2048 (exact)
    const int blocks   = subtiles / WAVES_PER_BLOCK; // 256

    ffq_wmma_kernel<<<blocks, 32 * WAVES_PER_BLOCK, 0, stream>>>(
        x, W1, b1, phi, W2, b2, out);
}